// MemoryNet_71021579206720
// MI455X (gfx1250) — compile-verified
//
#include <hip/hip_runtime.h>
#include <math.h>

typedef float v2f __attribute__((ext_vector_type(2)));
typedef float v8f __attribute__((ext_vector_type(8)));

#define BB    4
#define LL    2048
#define DD    128
#define EPSF  1e-12f

#define NCH   16            // K-chunks per batch for the q^T k GEMM
#define CHUNK (LL / NCH)    // 128 sequence rows per chunk
#define NRM   8             // L-chunks per batch for the norm reduction
#define LROW  (LL / NRM)    // 256 rows per norm block
#define PAD   132           // LDS row pitch (mult of 4, %64==4 -> conflict-free)

// ws layout (floats)
#define WS_NORM  0                        // BB*NRM*256
#define WS_PART  (BB * NRM * 256)         // BB*NCH*DD*DD
#define WS_SM    (WS_PART + BB * NCH * DD * DD)

// ---------------------------------------------------------------------------
// Kernel 1: partial sums of squares over L-chunks for q and k.
// grid = BB*NRM; threads 0..127 -> q channel d, 128..255 -> k channel d.
// ---------------------------------------------------------------------------
__global__ __launch_bounds__(256) void norm_part_kernel(const float* __restrict__ q,
                                                        const float* __restrict__ k,
                                                        float* __restrict__ normpart) {
    int b = blockIdx.x >> 3;
    int c = blockIdx.x & (NRM - 1);
    int t = threadIdx.x;
    int d = t & (DD - 1);
    const float* src = (t < DD) ? q : k;
    const float* p = src + ((size_t)b * LL + (size_t)c * LROW) * DD + d;
    float s = 0.f;
#pragma unroll 4
    for (int l = 0; l < LROW; ++l) {
        float x = p[(size_t)l * DD];
        s += x * x;
    }
    normpart[(size_t)blockIdx.x * 256 + t] = s;
}

// ---------------------------------------------------------------------------
// Kernel 2a: partial S = q^T k over one K-chunk of 128 rows, per batch.
// grid = BB*NCH blocks (8 waves each). q,k chunks staged K-major (transposed)
// in LDS so every WMMA operand is a single 8-byte ds load.
// ---------------------------------------------------------------------------
__global__ __launch_bounds__(256) void qk_part_kernel(const float* __restrict__ q,
                                                      const float* __restrict__ k,
                                                      float* __restrict__ partials) {
    __shared__ float qT[DD * PAD];   // qT[e][l] : 67.6 KB
    __shared__ float kT[DD * PAD];   // kT[e][l] : 67.6 KB
    int b = blockIdx.x >> 4;
    int c = blockIdx.x & (NCH - 1);
    int tid = threadIdx.x;

    const float* qb = q + ((size_t)b * LL + (size_t)c * CHUNK) * DD;
    const float* kb = k + ((size_t)b * LL + (size_t)c * CHUNK) * DD;

    // stage transposed: coalesced global reads, 4-bank-step LDS writes
    for (int i = tid; i < DD * CHUNK; i += 256) {
        int l = i >> 7, e = i & (DD - 1);
        qT[e * PAD + l] = qb[(size_t)l * DD + e];
        kT[e * PAD + l] = kb[(size_t)l * DD + e];
    }
    __syncthreads();

    int w   = tid >> 5;          // wave 0..7 owns d-strip [16w,16w+16)
    int lid = tid & 31;
    int g   = lid >> 4;
    int m   = lid & 15;
    int d_base = w * 16;

    v8f acc[8];
#pragma unroll
    for (int t = 0; t < 8; ++t) acc[t] = (v8f){0.f,0.f,0.f,0.f,0.f,0.f,0.f,0.f};

#pragma unroll 2
    for (int l0 = 0; l0 < CHUNK; l0 += 4) {
        // A[m][kk] = q[l0+kk, d_base+m] = qT[d_base+m][l0+kk], kk = 2g+j
        v2f a = *(const v2f*)&qT[(d_base + m) * PAD + l0 + 2 * g];
#pragma unroll
        for (int t = 0; t < 8; ++t) {
            // B[kk][n] = k[l0+kk, 16t+n] = kT[16t+n][l0+kk]
            v2f bv = *(const v2f*)&kT[(t * 16 + m) * PAD + l0 + 2 * g];
            acc[t] = __builtin_amdgcn_wmma_f32_16x16x4_f32(
                false, a, false, bv, (short)0, acc[t], false, false);
        }
    }

    float* Sp = partials + (size_t)blockIdx.x * DD * DD;
#pragma unroll
    for (int t = 0; t < 8; ++t)
#pragma unroll
        for (int r = 0; r < 8; ++r)
            Sp[(d_base + r + 8 * g) * DD + t * 16 + m] = acc[t][r];
}

// ---------------------------------------------------------------------------
// Kernel 2b: fold norm partials -> inverse norms, fold the NCH S-partials
// (ordered, deterministic), scale, row-softmax, write sm. grid = BB.
// ---------------------------------------------------------------------------
__global__ __launch_bounds__(256) void softmax_kernel(const float* __restrict__ normpart,
                                                      const float* __restrict__ partials,
                                                      float* __restrict__ sm) {
    __shared__ float nq[DD], nk[DD];
    __shared__ float Sb[DD * 129];   // padded rows -> conflict-free row softmax
    int b = blockIdx.x;
    int tid = threadIdx.x;

    if (tid < DD) {
        float sq = 0.f, sk = 0.f;
#pragma unroll
        for (int c = 0; c < NRM; ++c) {
            sq += normpart[(size_t)(b * NRM + c) * 256 + tid];
            sk += normpart[(size_t)(b * NRM + c) * 256 + DD + tid];
        }
        nq[tid] = 1.f / fmaxf(sqrtf(sq), EPSF);
        nk[tid] = 1.f / fmaxf(sqrtf(sk), EPSF);
    }
    __syncthreads();

    const float* Pb = partials + (size_t)b * NCH * DD * DD;
    for (int i = tid; i < DD * DD; i += 256) {
        float s = 0.f;
#pragma unroll
        for (int c = 0; c < NCH; ++c)
            s += Pb[(size_t)c * DD * DD + i];
        int d = i >> 7, e = i & (DD - 1);
        Sb[d * 129 + e] = s * nq[d] * nk[e];
    }
    __syncthreads();

    if (tid < DD) {
        int d = tid;
        float mx = -INFINITY;
        for (int e = 0; e < DD; ++e) mx = fmaxf(mx, Sb[d * 129 + e]);
        float sum = 0.f;
        for (int e = 0; e < DD; ++e) {
            float p = expf(Sb[d * 129 + e] - mx);
            Sb[d * 129 + e] = p;
            sum += p;
        }
        float r = 1.f / sum;
        for (int e = 0; e < DD; ++e)
            sm[((size_t)b * DD + d) * DD + e] = Sb[d * 129 + e] * r;
    }
}

// ---------------------------------------------------------------------------
// Kernel 3: out = v @ sm^T (M=2048, N=128, K=128 per batch) via WMMA.
// sm staged row-major with pad so the B operand is one ds_load_b64;
// A operand is K-contiguous in v -> one global_load_b64.
// ---------------------------------------------------------------------------
__global__ __launch_bounds__(256) void av_kernel(const float* __restrict__ v,
                                                 const float* __restrict__ sm,
                                                 float* __restrict__ out) {
    __shared__ float smL[DD * PAD];   // smL[d][e] : 67.6 KB
    int tid    = threadIdx.x;
    int gw     = blockIdx.x * 8 + (tid >> 5);   // global wave id 0..511
    int b      = gw >> 7;                       // all waves in a block share b
    int l_base = (gw & 127) * 16;

    const float* smb = sm + (size_t)b * DD * DD;
    for (int i4 = tid; i4 < (DD * DD) / 4; i4 += 256) {
        int d = i4 >> 5, e4 = i4 & 31;
        float4 val = ((const float4*)smb)[i4];
        *(float4*)&smL[d * PAD + e4 * 4] = val;
    }
    __syncthreads();

    int lid = tid & 31;
    int g   = lid >> 4;
    int m   = lid & 15;
    const float* vb = v + (size_t)b * LL * DD;

    v8f acc[8];
#pragma unroll
    for (int t = 0; t < 8; ++t) acc[t] = (v8f){0.f,0.f,0.f,0.f,0.f,0.f,0.f,0.f};

#pragma unroll 4
    for (int e0 = 0; e0 < DD; e0 += 4) {
        // A[m][kk] = v[l_base+m, e0+kk]  (K-contiguous in global memory)
        v2f a = *(const v2f*)&vb[(size_t)(l_base + m) * DD + e0 + 2 * g];
#pragma unroll
        for (int t = 0; t < 8; ++t) {
            // B[kk][n] = sm[16t+n, e0+kk] = smL[16t+n][e0+kk]
            v2f bv = *(const v2f*)&smL[(t * 16 + m) * PAD + e0 + 2 * g];
            acc[t] = __builtin_amdgcn_wmma_f32_16x16x4_f32(
                false, a, false, bv, (short)0, acc[t], false, false);
        }
    }

    float* ob = out + (size_t)b * LL * DD;
#pragma unroll
    for (int t = 0; t < 8; ++t)
#pragma unroll
        for (int r = 0; r < 8; ++r)
            ob[(size_t)(l_base + r + 8 * g) * DD + t * 16 + m] = acc[t][r];
}

// ---------------------------------------------------------------------------
extern "C" void kernel_launch(void* const* d_in, const int* in_sizes, int n_in,
                              void* d_out, int out_size, void* d_ws, size_t ws_size,
                              hipStream_t stream) {
    const float* q = (const float*)d_in[0];
    const float* k = (const float*)d_in[1];
    const float* v = (const float*)d_in[2];
    float* out = (float*)d_out;

    float* ws       = (float*)d_ws;
    float* normpart = ws + WS_NORM;   // BB*NRM*256 floats
    float* partials = ws + WS_PART;   // BB*NCH*DD*DD floats (4 MB)
    float* sm       = ws + WS_SM;     // BB*DD*DD floats

    norm_part_kernel<<<BB * NRM, 256, 0, stream>>>(q, k, normpart);
    qk_part_kernel<<<BB * NCH, 256, 0, stream>>>(q, k, partials);
    softmax_kernel<<<BB, 256, 0, stream>>>(normpart, partials, sm);
    av_kernel<<<(BB * (LL / 16)) / 8, 256, 0, stream>>>(v, sm, out);
}